// SSDLoss_56032143344093
// MI455X (gfx1250) — compile-verified
//
#include <hip/hip_runtime.h>

// SSD multibox loss for MI455X (gfx1250, wave32).
// Stage A uses V_WMMA_F32_16X16X4_F32 for the npc,ngc->npg cross-entropy GEMM.
// All WMMA operands are staged zero-padded in LDS so A/B loads are
// unconditional 8-byte ds_load_b64s (no divergent guards around WMMA).

#define NN 32
#define PP 8732
#define GG 24
#define CC 21
#define PRED_STRIDE 25
#define PTILE 16
#define NPT ((PP + PTILE - 1) / PTILE)   // 546
#define TB 256

typedef __attribute__((ext_vector_type(2))) float v2f;
typedef __attribute__((ext_vector_type(8))) float v8f;

__device__ __forceinline__ float smooth_l1(float d) {
    float ad = fabsf(d);
    return ad < 1.0f ? 0.5f * d * d : ad - 0.5f;
}

// ---------------------------------------------------------------------------
// Stage A: per (n, 16-anchor tile). One wave32 per block.
// Computes l_loc[n,p], l_conf[n,p], pos flag[n,p].
// ---------------------------------------------------------------------------
__global__ __launch_bounds__(32) void ssd_stageA(
    const float* __restrict__ pred,   // [N,P,25]
    const float* __restrict__ dflt,   // [P,4]
    const float* __restrict__ gt,     // [N,G,25]
    float* __restrict__ l_loc,        // [N,P]
    float* __restrict__ l_conf,       // [N,P]
    int*   __restrict__ posf)         // [N,P]
{
    // B-matrix: one-hot classes, zero-padded to [32 g][24 c], stride 24 (even)
    __shared__ __align__(16) float s_cls[32][24];
    // A-matrix: -log_softmax, zero-padded c -> 24, stride 24 (even)
    __shared__ __align__(16) float s_nlp[16][24];
    __shared__ float s_gtbox[GG][4];    // gt boxes for this sample
    __shared__ float s_df[16][4];       // default boxes for tile
    __shared__ float s_pb[16][4];       // predicted box regression for tile
    __shared__ float s_lp0[16];         // logp[:,0] per row (background)

    const int n    = blockIdx.y;
    const int p0   = blockIdx.x * PTILE;
    const int lane = threadIdx.x;       // 0..31

    // Stage zero-padded class matrix: s_cls[g][c] = gt[n,g,4+c] or 0.
    {
        const float* gbase = gt + (size_t)n * GG * 25;
        for (int idx = lane; idx < 32 * 24; idx += 32) {
            int g = idx / 24, c = idx % 24;
            float v = 0.0f;
            if (g < GG && c < CC) v = gbase[g * 25 + 4 + c];
            s_cls[g][c] = v;
        }
        // Stage gt boxes.
        for (int idx = lane; idx < GG * 4; idx += 32) {
            int g = idx >> 2, c = idx & 3;
            s_gtbox[g][c] = gbase[g * 25 + c];
        }
    }

    // Stage default boxes + predicted box regression (2 floats per lane).
    {
        int r = lane >> 1;              // row 0..15
        int h = lane & 1;               // which half of the 4 floats
        int p = p0 + r; if (p >= PP) p = PP - 1;
        s_df[r][2 * h + 0] = dflt[p * 4 + 2 * h + 0];
        s_df[r][2 * h + 1] = dflt[p * 4 + 2 * h + 1];
        const float* pr = pred + ((size_t)n * PP + p) * PRED_STRIDE;
        s_pb[r][2 * h + 0] = pr[2 * h + 0];
        s_pb[r][2 * h + 1] = pr[2 * h + 1];
    }

    // log-softmax per anchor row (lanes 0..15, one row each).
    if (lane < 16) {
        int p = p0 + lane; if (p >= PP) p = PP - 1;
        const float* row = pred + ((size_t)n * PP + p) * PRED_STRIDE + 4;
        float mx = row[0];
        for (int c = 1; c < CC; ++c) mx = fmaxf(mx, row[c]);
        float se = 0.0f;
        for (int c = 0; c < CC; ++c) se += expf(row[c] - mx);
        float lse = logf(se);
        for (int c = 0; c < CC; ++c) s_nlp[lane][c] = -(row[c] - mx - lse);
        for (int c = CC; c < 24; ++c) s_nlp[lane][c] = 0.0f;
        s_lp0[lane] = row[0] - mx - lse;
    }
    __syncthreads();

    // ----- WMMA: ce_pos[16 rows][24 gts] = (-logp) x one_hot^T ------------
    // A (16x4 f32): lane L holds M = L&15, K pair base (L>>4)*2.
    // B (4x16 f32): lane L holds N = L&15, same K pair base.
    // D (16x16 f32): lane L vgpr j -> M = j + 8*(L>>4), N = L&15.
    const int mrow  = lane & 15;
    const int khalf = (lane >> 4) * 2;  // 0 or 2

    v8f acc0 = {};   // g = 0..15
    v8f acc1 = {};   // g = 16..23 (24..31 are zero columns)
#pragma unroll
    for (int kk = 0; kk < 6; ++kk) {
        int c0 = kk * 4 + khalf;        // even, 0..22
        v2f a  = *(const v2f*)&s_nlp[mrow][c0];
        v2f b0 = *(const v2f*)&s_cls[mrow][c0];
        v2f b1 = *(const v2f*)&s_cls[16 + mrow][c0];
        acc0 = __builtin_amdgcn_wmma_f32_16x16x4_f32(false, a, false, b0,
                                                     (short)0, acc0, false, false);
        acc1 = __builtin_amdgcn_wmma_f32_16x16x4_f32(false, a, false, b1,
                                                     (short)0, acc1, false, false);
    }

    // ----- IoU match + masked reductions over g ---------------------------
#pragma unroll
    for (int j = 0; j < 8; ++j) {
        int row = j + ((lane >> 4) << 3);        // 0..15
        float dcx = s_df[row][0], dcy = s_df[row][1];
        float dw  = s_df[row][2], dh  = s_df[row][3];
        float px = s_pb[row][0], py = s_pb[row][1];
        float pw = s_pb[row][2], ph = s_pb[row][3];

        float sumc = 0.0f, suml = 0.0f; int cnt = 0;
#pragma unroll
        for (int t = 0; t < 2; ++t) {
            int g = (lane & 15) + 16 * t;
            if (g < GG) {
                float gcx = s_gtbox[g][0], gcy = s_gtbox[g][1];
                float gw  = s_gtbox[g][2], gh  = s_gtbox[g][3];
                float wi = fminf(gcx + gw * 0.5f, dcx + dw * 0.5f) -
                           fmaxf(gcx - gw * 0.5f, dcx - dw * 0.5f);
                wi = fmaxf(wi, 0.0f);
                float hi = fminf(gcy + gh * 0.5f, dcy + dh * 0.5f) -
                           fmaxf(gcy - gh * 0.5f, dcy - dh * 0.5f);
                hi = fmaxf(hi, 0.0f);
                float inter = wi * hi;
                float iou = inter / (gw * gh + dw * dh - inter);
                if (iou > 0.5f) {
                    float ce = (t == 0) ? acc0[j] : acc1[j];
                    float d0 = px - (gcx - dcx) / dw;
                    float d1 = py - (gcy - dcy) / dh;
                    float d2 = pw - logf(gw / dw);
                    float d3 = ph - logf(gh / dh);
                    suml += smooth_l1(d0) + smooth_l1(d1) +
                            smooth_l1(d2) + smooth_l1(d3);
                    sumc += ce;
                    cnt  += 1;
                }
            }
        }
        // butterfly sum across the 16-lane group (g dimension)
        for (int m = 1; m < 16; m <<= 1) {
            sumc += __shfl_xor(sumc, m, 32);
            suml += __shfl_xor(suml, m, 32);
            cnt  += __shfl_xor(cnt,  m, 32);
        }
        if ((lane & 15) == 0) {
            int p = p0 + row;
            if (p < PP) {
                size_t o = (size_t)n * PP + p;
                l_conf[o] = (cnt > 0) ? sumc : s_lp0[row];
                l_loc[o]  = suml;
                posf[o]   = (cnt > 0) ? 1 : 0;
            }
        }
    }
}

// ---------------------------------------------------------------------------
// Stage B: per-sample hard-negative mining via radix select + masked sum.
// ---------------------------------------------------------------------------
__device__ __forceinline__ unsigned f2o(float f) {
    unsigned u = __float_as_uint(f);
    return (u & 0x80000000u) ? ~u : (u | 0x80000000u);
}
__device__ __forceinline__ float o2f(unsigned o) {
    unsigned u = (o & 0x80000000u) ? (o & 0x7FFFFFFFu) : ~o;
    return __uint_as_float(u);
}

__global__ __launch_bounds__(TB) void ssd_stageB(
    const float* __restrict__ l_loc,
    const float* __restrict__ l_conf,
    const int*   __restrict__ posf,
    const int*   __restrict__ a_scalar,
    float* __restrict__ per_sample)
{
    __shared__ unsigned s_keys[PP];
    __shared__ float s_red[TB];
    __shared__ int   s_redi[TB];

    const int n   = blockIdx.x;
    const int tid = threadIdx.x;
    const float* lc = l_conf + (size_t)n * PP;
    const float* ll = l_loc  + (size_t)n * PP;
    const int*   pf = posf   + (size_t)n * PP;

    int myPos = 0;
    for (int p = tid; p < PP; p += TB) {
        s_keys[p] = f2o(lc[p]);
        myPos += pf[p];
    }
    s_redi[tid] = myPos; __syncthreads();
    for (int s = TB / 2; s > 0; s >>= 1) {
        if (tid < s) s_redi[tid] += s_redi[tid + s];
        __syncthreads();
    }
    int posRaw = s_redi[0]; __syncthreads();

    int negNum = min(PP - posRaw, 3 * posRaw);
    int posNum = max(posRaw, 1);
    negNum = max(negNum, 1);

    auto selectK = [&](int k) -> unsigned {
        unsigned prefix = 0u;
        for (int bit = 31; bit >= 0; --bit) {
            int cnt = 0;
            for (int p = tid; p < PP; p += TB) {
                unsigned key = s_keys[p];
                bool pm = (bit == 31) || ((key >> (bit + 1)) == prefix);
                if (pm && (((key >> bit) & 1u) == 0u)) cnt++;
            }
            s_redi[tid] = cnt; __syncthreads();
            for (int s = TB / 2; s > 0; s >>= 1) {
                if (tid < s) s_redi[tid] += s_redi[tid + s];
                __syncthreads();
            }
            int czero = s_redi[0]; __syncthreads();
            if (k < czero) { prefix = prefix << 1; }
            else           { k -= czero; prefix = (prefix << 1) | 1u; }
        }
        return prefix;
    };

    unsigned thrNegU = selectK(negNum - 1);      // neg_num-th smallest
    unsigned thrPosU = selectK(PP - posNum);     // pos_num-th largest
    float thrNeg = o2f(thrNegU);
    float thrPos = o2f(thrPosU);
    float aval = (float)a_scalar[0];

    float acc = 0.0f;
    for (int p = tid; p < PP; p += TB) {
        float c = lc[p];
        bool valid = (thrNeg > c) || (thrPos < c);
        if (valid) acc += ll[p] + aval * fabsf(c);
    }
    s_red[tid] = acc; __syncthreads();
    for (int s = TB / 2; s > 0; s >>= 1) {
        if (tid < s) s_red[tid] += s_red[tid + s];
        __syncthreads();
    }
    if (tid == 0) per_sample[n] = s_red[0] / (float)posNum;
}

// ---------------------------------------------------------------------------
// Stage C: mean over samples.
// ---------------------------------------------------------------------------
__global__ __launch_bounds__(32) void ssd_stageC(
    const float* __restrict__ per_sample, float* __restrict__ out)
{
    int lane = threadIdx.x;
    float v = (lane < NN) ? per_sample[lane] : 0.0f;
    for (int m = 16; m >= 1; m >>= 1) v += __shfl_xor(v, m, 32);
    if (lane == 0) out[0] = v / (float)NN;
}

extern "C" void kernel_launch(void* const* d_in, const int* in_sizes, int n_in,
                              void* d_out, int out_size, void* d_ws, size_t ws_size,
                              hipStream_t stream) {
    const float* pred = (const float*)d_in[0];   // [N,P,25]
    const float* dflt = (const float*)d_in[1];   // [P,4]
    const float* gt   = (const float*)d_in[2];   // [N,G,25]
    const int*   a    = (const int*)d_in[3];     // scalar
    float* out = (float*)d_out;

    float* l_loc      = (float*)d_ws;                        // N*P
    float* l_conf     = l_loc + (size_t)NN * PP;             // N*P
    int*   posf       = (int*)(l_conf + (size_t)NN * PP);    // N*P
    float* per_sample = (float*)(posf + (size_t)NN * PP);    // N

    dim3 gA(NPT, NN);
    ssd_stageA<<<gA, 32, 0, stream>>>(pred, dflt, gt, l_loc, l_conf, posf);
    ssd_stageB<<<NN, TB, 0, stream>>>(l_loc, l_conf, posf, a, per_sample);
    ssd_stageC<<<1, 32, 0, stream>>>(per_sample, out);
}